// EigenDepth_75651553951791
// MI455X (gfx1250) — compile-verified
//
#include <hip/hip_runtime.h>
#include <hip/hip_bf16.h>
#include <math.h>

#define IMG_H   384
#define IMG_W   128
#define GH      6
#define GW      2
#define PH      64
#define PW      64
#define NPIX    4096        // PH*PW
#define FEAT_DIM 72         // GH*GW*6
#define NCLS    751
#define NBATCH  256

typedef float v2f __attribute__((ext_vector_type(2)));
typedef float v8f __attribute__((ext_vector_type(8)));

// ---------------------------------------------------------------------------
// Kernel 1: per-patch feature moments -> covariance -> 6x6 Jacobi eigvals ->
//           log-eigen features.  One workgroup (256 thr = 8 waves) per patch.
// Memory-bound: one ~50MB pass over z at HBM rate; everything else in regs.
// ---------------------------------------------------------------------------
__global__ __launch_bounds__(256)
void cov_eig_kernel(const float* __restrict__ z, float* __restrict__ E)
{
    const int patch = blockIdx.x;          // 0..11  (gy*GW + gx)
    const int b     = blockIdx.y;          // 0..255
    const int gy = patch >> 1;
    const int gx = patch & 1;
    const float* zb = z + (size_t)b * IMG_H * IMG_W;
    const int tid = threadIdx.x;

    float s[6];
    float q[21];
#pragma unroll
    for (int i = 0; i < 6; ++i)  s[i] = 0.f;
#pragma unroll
    for (int i = 0; i < 21; ++i) q[i] = 0.f;

    // 16 pixels per thread, coalesced 64-wide row segments
#pragma unroll
    for (int it = 0; it < NPIX / 256; ++it) {
        const int pix = tid + it * 256;
        const int lr = pix >> 6;           // local row 0..63
        const int lc = pix & 63;           // local col 0..63
        const int h = gy * PH + lr;
        const int w = gx * PW + lc;
        const float zc = zb[h * IMG_W + w];
        const float dzdx = (w + 1 < IMG_W) ? (zb[h * IMG_W + w + 1] - zc) : 0.f;
        const float dzdy = (h + 1 < IMG_H) ? (zb[(h + 1) * IMG_W + w] - zc) : 0.f;

        const float nx = -dzdx, ny = -dzdy;
        const float inv = 1.0f / sqrtf(nx * nx + ny * ny + 1.0f + 1e-6f);

        float f[6];
        f[0] = (float)w * (1.0f / IMG_W);
        f[1] = (float)h * (1.0f / IMG_H);
        f[2] = zc;
        f[3] = nx * inv;
        f[4] = ny * inv;
        f[5] = inv;            // n_z * inv_norm with n_z = 1

        int idx = 0;
#pragma unroll
        for (int c = 0; c < 6; ++c) {
            s[c] += f[c];
#pragma unroll
            for (int d = c; d < 6; ++d)
                q[idx++] += f[c] * f[d];
        }
    }

    // wave32 butterfly reduction
#pragma unroll
    for (int i = 0; i < 6; ++i)
        for (int off = 16; off > 0; off >>= 1)
            s[i] += __shfl_down(s[i], off, 32);
#pragma unroll
    for (int i = 0; i < 21; ++i)
        for (int off = 16; off > 0; off >>= 1)
            q[i] += __shfl_down(q[i], off, 32);

    __shared__ float red[8][27];
    const int wave = tid >> 5;
    const int lane = tid & 31;
    if (lane == 0) {
#pragma unroll
        for (int i = 0; i < 6; ++i)  red[wave][i]     = s[i];
#pragma unroll
        for (int i = 0; i < 21; ++i) red[wave][6 + i] = q[i];
    }
    __syncthreads();

    if (tid == 0) {
        float S[6], Q[21];
        for (int i = 0; i < 6; ++i)  { float a = 0.f; for (int w2 = 0; w2 < 8; ++w2) a += red[w2][i];     S[i] = a; }
        for (int i = 0; i < 21; ++i) { float a = 0.f; for (int w2 = 0; w2 < 8; ++w2) a += red[w2][6 + i]; Q[i] = a; }

        float A[6][6];
        const float invN   = 1.0f / (float)NPIX;
        const float invNm1 = 1.0f / (float)(NPIX - 1);
        int idx = 0;
        for (int c = 0; c < 6; ++c)
            for (int d = c; d < 6; ++d) {
                float v = (Q[idx] - S[c] * S[d] * invN) * invNm1;
                if (c == d) v += 1e-5f;
                A[c][d] = v;
                A[d][c] = v;
                ++idx;
            }

        // cyclic Jacobi: 10 sweeps is ample for 6x6
        for (int sweep = 0; sweep < 10; ++sweep) {
            for (int p = 0; p < 5; ++p)
                for (int qi = p + 1; qi < 6; ++qi) {
                    const float apq = A[p][qi];
                    if (fabsf(apq) < 1e-13f) continue;
                    const float theta = (A[qi][qi] - A[p][p]) / (2.0f * apq);
                    const float t = copysignf(1.0f, theta) /
                                    (fabsf(theta) + sqrtf(theta * theta + 1.0f));
                    const float c  = 1.0f / sqrtf(t * t + 1.0f);
                    const float sn = t * c;
                    for (int k = 0; k < 6; ++k) {        // A <- A * G
                        const float akp = A[k][p], akq = A[k][qi];
                        A[k][p]  = c * akp - sn * akq;
                        A[k][qi] = sn * akp + c * akq;
                    }
                    for (int k = 0; k < 6; ++k) {        // A <- G^T * A
                        const float apk = A[p][k], aqk = A[qi][k];
                        A[p][k]  = c * apk - sn * aqk;
                        A[qi][k] = sn * apk + c * aqk;
                    }
                }
        }

        float e[6];
        for (int i = 0; i < 6; ++i) e[i] = A[i][i];
        // ascending sort (eigvalsh order)
        for (int i = 0; i < 5; ++i)
            for (int j = 0; j < 5 - i; ++j)
                if (e[j] > e[j + 1]) { float tt = e[j]; e[j] = e[j + 1]; e[j + 1] = tt; }

        float* dst = E + (size_t)b * FEAT_DIM + patch * 6;
        for (int i = 0; i < 6; ++i) dst[i] = logf(e[i] + 1e-6f);
    }
}

// ---------------------------------------------------------------------------
// Kernel 2: out[256,751] = E[256,72] @ W_cls^T[72,751] + b  via
//           V_WMMA_F32_16X16X4_F32.  One wave per 16x16 output tile.
//           M tiles = 16, N tiles = 47 -> 752 waves = 94 blocks x 8 waves.
// A lane layout (16x4 f32): M = lane%16, K = (lane/16)*2 + {0,1}
// B lane layout (4x16 f32): N = lane%16, K = (lane/16)*2 + {0,1}
// D layout: vgpr r -> row r + 8*(lane/16), col lane%16
// ---------------------------------------------------------------------------
__global__ __launch_bounds__(256)
void cls_gemm_wmma(const float* __restrict__ E, const float* __restrict__ Wc,
                   const float* __restrict__ bc, float* __restrict__ out)
{
    const int lane = threadIdx.x & 31;
    const int wave = threadIdx.x >> 5;
    const int tile = blockIdx.x * 8 + wave;     // 0..751
    const int mT = tile / 47;
    const int nT = tile % 47;
    const int m0 = mT * 16;
    const int n0 = nT * 16;

    const int lm = lane & 15;
    const int lk = (lane >> 4) << 1;            // 0 or 2

    const int n  = n0 + lm;
    const int nc = (n < NCLS) ? n : (NCLS - 1); // clamp: keep EXEC all-1s for WMMA

    const float* Erow = E  + (size_t)(m0 + lm) * FEAT_DIM;
    const float* Wrow = Wc + (size_t)nc * FEAT_DIM;

    v8f acc = {};
#pragma unroll
    for (int k0 = 0; k0 < FEAT_DIM; k0 += 4) {
        v2f a, bm;
        a.x  = Erow[k0 + lk];
        a.y  = Erow[k0 + lk + 1];
        bm.x = Wrow[k0 + lk];
        bm.y = Wrow[k0 + lk + 1];
        acc = __builtin_amdgcn_wmma_f32_16x16x4_f32(
                  /*neg_a=*/false, a, /*neg_b=*/false, bm,
                  /*c_mod=*/(short)0, acc, /*reuse_a=*/false, /*reuse_b=*/false);
    }

    if (n < NCLS) {
        const float bias  = bc[n];
        const int   rbase = m0 + ((lane >> 4) << 3);
#pragma unroll
        for (int r = 0; r < 8; ++r)
            out[(size_t)(rbase + r) * NCLS + n] = acc[r] + bias;
    }
}

// ---------------------------------------------------------------------------
extern "C" void kernel_launch(void* const* d_in, const int* in_sizes, int n_in,
                              void* d_out, int out_size, void* d_ws, size_t ws_size,
                              hipStream_t stream)
{
    const float* x  = (const float*)d_in[0];   // (256,1,384,128) f32 -> z
    const float* Wc = (const float*)d_in[1];   // (751,72) f32
    const float* bc = (const float*)d_in[2];   // (751,) f32
    float* out = (float*)d_out;                // (256,751) f32
    float* E   = (float*)d_ws;                 // (256,72) f32 eigen-features

    dim3 g1(GH * GW, NBATCH);                  // 12 patches x 256 images
    cov_eig_kernel<<<g1, 256, 0, stream>>>(x, E);

    cls_gemm_wmma<<<94, 256, 0, stream>>>(E, Wc, bc, out);
}